// ParallelNoise_51436528337176
// MI455X (gfx1250) — compile-verified
//
#include <hip/hip_runtime.h>
#include <cstdint>

// Problem constants (from reference: B=32, C=3, H=W=224, NUM=16, SIGMA=0.1)
#define BATCH  32
#define NUMT   16
#define CHW    150528            // 3*224*224
#define CHW4   (CHW / 4)         // 37632 float4 elements
#define TILES  (CHW / 16)        // 9408 WMMA output tiles per image
#define SIGMA  0.1f
#define EPSV   1e-20f

#define WTILE  6272              // floats per TDM tile of w (25088 B); CHW/WTILE = 24
#define WTILE4 (WTILE / 4)

typedef __attribute__((ext_vector_type(2))) float v2f;
typedef __attribute__((ext_vector_type(8))) float v8f;
typedef __attribute__((ext_vector_type(4))) unsigned int u32x4;
typedef __attribute__((ext_vector_type(4))) int i32x4;
typedef __attribute__((ext_vector_type(8))) int i32x8;

// ---------------------------------------------------------------------------
// Counter-based PRNG: lets every pass (and every WMMA A-fragment lane slot)
// regenerate t[n,b,d] on demand instead of spilling a 308MB tensor to HBM.
// ---------------------------------------------------------------------------
__device__ __forceinline__ uint32_t hashu(uint32_t x) {
    x ^= x >> 16; x *= 0x7feb352dU;
    x ^= x >> 15; x *= 0x846ca68bU;
    x ^= x >> 16; return x;
}

// Approximate standard normal (Irwin-Hall, 4 uniforms). The reference
// standardizes each (n,b) sample to zero-mean / sigma-std anyway.
__device__ __forceinline__ float zgen(uint32_t n, uint32_t b, uint32_t d) {
    uint32_t idx = (n * (uint32_t)BATCH + b) * (uint32_t)CHW + d;
    uint32_t h0 = hashu(idx ^ 0x9E3779B9u);
    uint32_t h1 = hashu(idx + 0x85EBCA6Bu);
    float u0 = (float)(h0 & 0xFFFFu);
    float u1 = (float)(h0 >> 16);
    float u2 = (float)(h1 & 0xFFFFu);
    float u3 = (float)(h1 >> 16);
    return ((u0 + u1 + u2 + u3) * (1.0f / 65536.0f) - 2.0f) * 1.7320508f;
}

__device__ __forceinline__ float clipf(float x) {
    return fminf(fmaxf(x, 0.0f), 1.0f);
}

// Block-wide sum, wave32 aware (8 waves / 256-thread block). Result in tid 0.
__device__ __forceinline__ float block_sum(float v, float* sh) {
    const int tid = threadIdx.x;
#pragma unroll
    for (int off = 16; off > 0; off >>= 1) v += __shfl_down(v, off, 32);
    __syncthreads();                         // protect sh reuse across calls
    if ((tid & 31) == 0) sh[tid >> 5] = v;
    __syncthreads();
    v = (tid < (int)(blockDim.x >> 5)) ? sh[tid] : 0.0f;
#pragma unroll
    for (int off = 16; off > 0; off >>= 1) v += __shfl_down(v, off, 32);
    return v;
}

// ---------------------------------------------------------------------------
// Tensor Data Mover: DMA a contiguous run of `nelem` f32 (nelem < 65536,
// multiple of 4B) from global into LDS. Builds the D# per CDNA5 ISA
// §8.3/8.4: group0 = {count=1, lds_addr, global_addr, type=2}; group1 =
// {data_size=4B, tensor_dim0 = tile_dim0 = nelem, 1 row}. Descriptor values
// are wave-uniform -> SGPRs. Caller must be a full wave; TENSORcnt waited
// here, LDS visibility to other waves needs a barrier outside.
// ---------------------------------------------------------------------------
__device__ __forceinline__ void tdm_load_1d(const float* gsrc, float* ldsdst,
                                            unsigned int nelem) {
    uint64_t ga = (uint64_t)(uintptr_t)gsrc;
    uint32_t la = (uint32_t)(uintptr_t)ldsdst;   // low 32 bits = LDS byte offset
    u32x4 g0;
    g0[0] = 1u;                                   // count=1 valid descriptor
    g0[1] = la;                                   // lds_addr [63:32]
    g0[2] = (uint32_t)ga;                         // global_addr lo [95:64]
    g0[3] = (uint32_t)((ga >> 32) & 0x01FFFFFFu)  // global_addr hi [120:96]
            | 0x80000000u;                        // type=2 ("image") [127:126]
    i32x8 g1;
    g1[0] = (int)0x00020000u;                     // data_size=2 (4 bytes)
    g1[1] = (int)((nelem & 0xFFFFu) << 16);       // tensor_dim0 lo16 @ [63:48]
    g1[2] = (int)((nelem >> 16) | (1u << 16));    // tensor_dim0 hi16, tensor_dim1=1
    g1[3] = (int)((nelem & 0xFFFFu) << 16);       // tile_dim0 @ [127:112]
    g1[4] = 1;                                    // tile_dim1=1, tile_dim2=0
    g1[5] = (int)nelem;                           // tensor_dim0_stride lo32
    g1[6] = 0;
    g1[7] = 0;
#if __has_include(<hip/amd_detail/amd_gfx1250_TDM.h>)
    // amdgpu-toolchain (therock headers) -> 6-arg builtin
    __builtin_amdgcn_tensor_load_to_lds(g0, g1, (i32x4)0, (i32x4)0, (i32x8)0, 0);
#else
    // ROCm 7.2 clang-22 -> 5-arg builtin
    __builtin_amdgcn_tensor_load_to_lds(g0, g1, (i32x4)0, (i32x4)0, 0);
#endif
    __builtin_amdgcn_s_wait_tensorcnt(0);
}

// ---------------------------------------------------------------------------
// Pass 1: per-(n,b) mean / SIGMA*rsqrt(var) of the raw noise stream.
// Pure PRNG + reduction, no memory reads.
// ---------------------------------------------------------------------------
__global__ __launch_bounds__(256) void k_stats(float* __restrict__ mean,
                                               float* __restrict__ scale) {
    __shared__ float sh[8];
    const int nb = blockIdx.x;
    const int n = nb >> 5;          // nb / 32
    const int b = nb & 31;
    float s = 0.0f, ss = 0.0f;
    for (int d = threadIdx.x; d < CHW; d += blockDim.x) {
        float z = zgen(n, b, d);
        s += z;
        ss = fmaf(z, z, ss);
    }
    float ts  = block_sum(s, sh);
    float tss = block_sum(ss, sh);
    if (threadIdx.x == 0) {
        float m   = ts * (1.0f / (float)CHW);
        float var = (tss - ts * m) * (1.0f / (float)(CHW - 1));  // unbiased
        mean[nb]  = m;
        scale[nb] = SIGMA * rsqrtf(var);
    }
}

// ---------------------------------------------------------------------------
// Pass 2a: prob2[b] = sigmoid(clip(imgs+noise) . w + b0)
// ---------------------------------------------------------------------------
__global__ __launch_bounds__(256) void k_prob2(const float4* __restrict__ imgs,
                                               const float4* __restrict__ noise,
                                               const float4* __restrict__ w,
                                               const float*  __restrict__ bias,
                                               float* __restrict__ prob2) {
    __shared__ float sh[8];
    const int b = blockIdx.x;
    const float4* ip = imgs  + (size_t)b * CHW4;
    const float4* np = noise + (size_t)b * CHW4;
    float acc = 0.0f;
    for (int d4 = threadIdx.x; d4 < CHW4; d4 += blockDim.x) {
        __builtin_prefetch((const void*)(ip + d4 + 2 * blockDim.x), 0, 1);
        float4 iv = ip[d4], nv = np[d4], wv = w[d4];
        acc = fmaf(clipf(iv.x + nv.x), wv.x, acc);
        acc = fmaf(clipf(iv.y + nv.y), wv.y, acc);
        acc = fmaf(clipf(iv.z + nv.z), wv.z, acc);
        acc = fmaf(clipf(iv.w + nv.w), wv.w, acc);
    }
    float t = block_sum(acc, sh);
    if (threadIdx.x == 0)
        prob2[b] = 1.0f / (1.0f + expf(-(t + bias[0])));
}

// ---------------------------------------------------------------------------
// Pass 2b: diff[n,b] = prob2[b] - sigmoid(clip(imgs+noise+t) . w + b0)
// Heaviest pass (512 blocks). w is staged through LDS via the Tensor Data
// Mover in 24 tiles of 25KB: wave 0 issues the DMA + s_wait_tensorcnt, a
// block barrier publishes it; the DMA overlaps the PRNG-bound VALU stream
// and removes the per-lane w traffic from L2. imgs/noise stay as coalesced
// float4 loads (L2-resident, 38MB << 192MB L2).
// ---------------------------------------------------------------------------
__global__ __launch_bounds__(256) void k_diff(const float4* __restrict__ imgs,
                                              const float4* __restrict__ noise,
                                              const float*  __restrict__ w,
                                              const float*  __restrict__ bias,
                                              const float*  __restrict__ mean,
                                              const float*  __restrict__ scale,
                                              const float*  __restrict__ prob2,
                                              float* __restrict__ diff) {
    __shared__ float sh[8];
    __shared__ __align__(16) float wtile[WTILE];
    const int nb = blockIdx.x;
    const int n = nb >> 5;
    const int b = nb & 31;
    const float m  = mean[nb];
    const float sc = scale[nb];
    const float4* ip = imgs  + (size_t)b * CHW4;
    const float4* np = noise + (size_t)b * CHW4;
    float acc = 0.0f;
    for (int base = 0; base < CHW; base += WTILE) {
        if (threadIdx.x < 32)                       // wave 0 issues the DMA
            tdm_load_1d(w + base, wtile, WTILE);
        __syncthreads();                            // publish LDS tile
        const float4* wt = (const float4*)wtile;
        const int c4 = base >> 2;
        for (int j = threadIdx.x; j < WTILE4; j += blockDim.x) {
            const int d4 = c4 + j;
            __builtin_prefetch((const void*)(ip + d4 + 2 * blockDim.x), 0, 1);
            float4 iv = ip[d4], nv = np[d4], wv = wt[j];
            int d = d4 << 2;
            float t0 = sc * (zgen(n, b, d + 0) - m);
            float t1 = sc * (zgen(n, b, d + 1) - m);
            float t2 = sc * (zgen(n, b, d + 2) - m);
            float t3 = sc * (zgen(n, b, d + 3) - m);
            acc = fmaf(clipf(iv.x + nv.x + t0), wv.x, acc);
            acc = fmaf(clipf(iv.y + nv.y + t1), wv.y, acc);
            acc = fmaf(clipf(iv.z + nv.z + t2), wv.z, acc);
            acc = fmaf(clipf(iv.w + nv.w + t3), wv.w, acc);
        }
        __syncthreads();                            // tile reuse guard
    }
    float tsum = block_sum(acc, sh);
    if (threadIdx.x == 0) {
        float ap = 1.0f / (1.0f + expf(-(tsum + bias[0])));
        diff[nb] = prob2[b] - ap;
    }
}

// ---------------------------------------------------------------------------
// Pass 3: out[b,d] = sum_n diff'[n,b] * t[n,b,d]   via V_WMMA_F32_16X16X4_F32
//   M rows  = 16 output elements d (tile)
//   K       = trial index n, 4 per WMMA, 4 chained WMMAs cover NUM=16
//   B frag  = diff[n,b] / ((||diff||+eps)*SIGMA), broadcast across N columns
// A-fragment layout (ISA 7.12.2): lane L<16 holds A[M=L, K={r,r+1}],
// lanes 16-31 hold K={r+2,r+3}. Counter-based PRNG -> each lane generates
// exactly its slot's (n,d) values, no shuffles. All N columns of C are equal;
// lanes with (lane&15)==0 store rows M=0..7 / 8..15 as two b128 stores.
// ---------------------------------------------------------------------------
__global__ __launch_bounds__(256) void k_out(const float* __restrict__ mean,
                                             const float* __restrict__ scale,
                                             const float* __restrict__ diff,
                                             float* __restrict__ out) {
    const int b    = blockIdx.y;
    const int lane = threadIdx.x & 31;
    const int wave = threadIdx.x >> 5;
    const int wavesTotal = gridDim.x * (blockDim.x >> 5);
    const int waveId     = blockIdx.x * (blockDim.x >> 5) + wave;

    // Per-image scalars (uniform across the wave -> scalar loads)
    float dp[NUMT], mn[NUMT], sc[NUMT];
    float s2 = 0.0f;
#pragma unroll
    for (int n = 0; n < NUMT; ++n) {
        float df = diff[n * BATCH + b];
        dp[n] = df;
        s2 = fmaf(df, df, s2);
        mn[n] = mean[n * BATCH + b];
        sc[n] = scale[n * BATCH + b];
    }
    const float inv = 1.0f / ((sqrtf(s2) + EPSV) * SIGMA);
#pragma unroll
    for (int n = 0; n < NUMT; ++n) dp[n] *= inv;

    const int h = lane >> 4;       // lane half selects K sub-pair
    const int m = lane & 15;       // A-matrix row -> output element in tile

    for (int tile = waveId; tile < TILES; tile += wavesTotal) {
        const int d0 = tile * 16;
        const uint32_t dd = (uint32_t)(d0 + m);
        v8f c = {0.0f, 0.0f, 0.0f, 0.0f, 0.0f, 0.0f, 0.0f, 0.0f};
#pragma unroll
        for (int i = 0; i < 4; ++i) {
            const int kx = i * 4 + 2 * h;  // K slots {kx, kx+1} for this lane
            v2f a, bf;
            a.x  = sc[kx    ] * (zgen((uint32_t)(kx    ), (uint32_t)b, dd) - mn[kx    ]);
            a.y  = sc[kx + 1] * (zgen((uint32_t)(kx + 1), (uint32_t)b, dd) - mn[kx + 1]);
            bf.x = dp[kx];
            bf.y = dp[kx + 1];
            // D = A(16x4 f32) x B(4x16 f32) + C, full f32 precision
            c = __builtin_amdgcn_wmma_f32_16x16x4_f32(
                    /*neg_a=*/false, a, /*neg_b=*/false, bf,
                    /*c_mod=*/(short)0, c, /*reuse_a=*/false, /*reuse_b=*/false);
        }
        // C layout: lanes 0-15 -> M=r (N=lane), lanes 16-31 -> M=r+8.
        // Columns are identical; N==0 lanes of each half store 8 rows.
        if (m == 0) {
            float* op = out + (size_t)b * CHW + d0 + h * 8;
            float4 lo = make_float4(c[0], c[1], c[2], c[3]);
            float4 hi = make_float4(c[4], c[5], c[6], c[7]);
            *(float4*)(op)     = lo;
            *(float4*)(op + 4) = hi;
        }
    }
}

// ---------------------------------------------------------------------------
extern "C" void kernel_launch(void* const* d_in, const int* in_sizes, int n_in,
                              void* d_out, int out_size, void* d_ws, size_t ws_size,
                              hipStream_t stream) {
    const float* imgs  = (const float*)d_in[0];
    const float* noise = (const float*)d_in[1];
    const float* w     = (const float*)d_in[2];
    const float* bias  = (const float*)d_in[3];
    float* out = (float*)d_out;

    // Workspace: mean[512] | scale[512] | prob2[32] | diff[512]  (6.2 KB)
    float* ws    = (float*)d_ws;
    float* mean  = ws;
    float* scale = ws + NUMT * BATCH;
    float* prob2 = ws + 2 * NUMT * BATCH;
    float* diffp = ws + 2 * NUMT * BATCH + BATCH;

    k_stats<<<NUMT * BATCH, 256, 0, stream>>>(mean, scale);
    k_prob2<<<BATCH, 256, 0, stream>>>((const float4*)imgs, (const float4*)noise,
                                       (const float4*)w, bias, prob2);
    k_diff<<<NUMT * BATCH, 256, 0, stream>>>((const float4*)imgs, (const float4*)noise,
                                             w, bias, mean, scale, prob2, diffp);
    dim3 g3(147, BATCH);   // 147 blocks * 8 waves = 1176 waves per image; 9408 tiles
    k_out<<<g3, 256, 0, stream>>>(mean, scale, diffp, out);
}